// KGNN_11089605558979
// MI455X (gfx1250) — compile-verified
//
#include <hip/hip_runtime.h>
#include <math.h>

typedef __attribute__((ext_vector_type(2))) float v2f;
typedef __attribute__((ext_vector_type(8))) float v8f;

#define WMMA_F32X4(a, b, c) \
    __builtin_amdgcn_wmma_f32_16x16x4_f32(false, (a), false, (b), (short)0, (c), false, false)

// -------------------------------------------------------------------------
// Edge-gate MLP: g[e] = sigmoid( relu(ea[e,0:16] @ mw1[16,128] + b1) @ mw2 + b2 )
// One wave handles 16 edges. A = 16x16 edge_attr tile (four K=4 WMMA steps),
// B = mw1 column tiles (8 tiles of 16). Hidden activations stay in the C
// fragment; relu + dot-with-mw2 accumulated per lane, then reduced across the
// 16-lane halves with shfl_xor.
// -------------------------------------------------------------------------
__global__ __launch_bounds__(256) void pdn_gate_kernel(
    const float* __restrict__ ea, const float* __restrict__ mw1,
    const float* __restrict__ mb1, const float* __restrict__ mw2,
    const float* __restrict__ mb2, float* __restrict__ g, int E) {
  const int wave = threadIdx.x >> 5;
  const int lane = threadIdx.x & 31;
  const int half = lane >> 4;   // 0: lanes 0-15, 1: lanes 16-31
  const int l16  = lane & 15;
  const int e0 = (blockIdx.x * 8 + wave) * 16;
  if (e0 >= E) return;  // wave-uniform exit, EXEC stays all-1s for WMMA

  int er = e0 + l16;
  if (er >= E) er = E - 1;  // clamped load (store is guarded)
  const float* earow = ea + (size_t)er * 16;

  // A fragments for the 4 K-steps: VGPR0 holds K = 4*kk + 2*half, VGPR1 = +1
  v2f afrag[4];
#pragma unroll
  for (int kk = 0; kk < 4; ++kk) {
    afrag[kk].x = earow[4 * kk + 2 * half];
    afrag[kk].y = earow[4 * kk + 2 * half + 1];
  }

  float acc[8];
#pragma unroll
  for (int v = 0; v < 8; ++v) acc[v] = 0.0f;

#pragma unroll
  for (int t = 0; t < 8; ++t) {  // 8 column tiles of the 128-wide hidden layer
    const int n0 = t * 16;
    v8f c = {};
#pragma unroll
    for (int kk = 0; kk < 4; ++kk) {
      const int kb = 4 * kk + 2 * half;
      v2f b;
      b.x = mw1[(size_t)kb * 128 + n0 + l16];
      b.y = mw1[(size_t)(kb + 1) * 128 + n0 + l16];
      c = WMMA_F32X4(afrag[kk], b, c);
    }
    const float b1 = mb1[n0 + l16];
    const float w2 = mw2[n0 + l16];
#pragma unroll
    for (int v = 0; v < 8; ++v) {
      // c[v] is hidden col (n0+l16) for edge row (v + 8*half)
      acc[v] += fmaxf(c[v] + b1, 0.0f) * w2;
    }
  }

  // Reduce across the 16 lanes of each half (masks 1,2,4,8 stay in-half)
#pragma unroll
  for (int off = 1; off < 16; off <<= 1) {
#pragma unroll
    for (int v = 0; v < 8; ++v) acc[v] += __shfl_xor(acc[v], off, 32);
  }

  const float b2 = mb2[0];
#pragma unroll
  for (int v = 0; v < 8; ++v) {
    if (l16 == v) {
      const int e = e0 + v + 8 * half;
      if (e < E) {
        const float s = acc[v] + b2;
        g[e] = 1.0f / (1.0f + __expf(-s));
      }
    }
  }
}

// -------------------------------------------------------------------------
// h[M,128] = act(x[M,128]) @ W[128,128], act = identity or relu (fused).
// 256 threads = 8 waves; wave w owns column tile [16w,16w+16) and computes
// RPW=4 row tiles (64 rows per block), reusing each B (W) fragment across
// 4 WMMAs -> 4x fewer W load instructions, 128 WMMAs per wave.
// -------------------------------------------------------------------------
#define RPW 4
__global__ __launch_bounds__(256) void pdn_gemm_kernel(
    const float* __restrict__ x, const float* __restrict__ W,
    float* __restrict__ h, int M, int relu_in) {
  const int wave = threadIdx.x >> 5;
  const int lane = threadIdx.x & 31;
  const int half = lane >> 4;
  const int l16  = lane & 15;
  const int n0 = wave * 16;
  const int rowBase = blockIdx.x * (16 * RPW);

  const float* xrow[RPW];
#pragma unroll
  for (int t = 0; t < RPW; ++t) {
    int ar = rowBase + t * 16 + l16;
    if (ar >= M) ar = M - 1;  // clamped loads; stores guarded below
    xrow[t] = x + (size_t)ar * 128;
  }

  v8f c[RPW];
#pragma unroll
  for (int t = 0; t < RPW; ++t) c[t] = (v8f){};

#pragma unroll 4
  for (int k0 = 0; k0 < 128; k0 += 4) {
    const int ka = k0 + half * 2;
    v2f b;  // B fragment loaded once, reused by all row tiles
    b.x = W[(size_t)ka * 128 + n0 + l16];
    b.y = W[(size_t)(ka + 1) * 128 + n0 + l16];
#pragma unroll
    for (int t = 0; t < RPW; ++t) {
      v2f a;
      a.x = xrow[t][ka];
      a.y = xrow[t][ka + 1];
      if (relu_in) { a.x = fmaxf(a.x, 0.0f); a.y = fmaxf(a.y, 0.0f); }
      c[t] = WMMA_F32X4(a, b, c[t]);
    }
  }

#pragma unroll
  for (int t = 0; t < RPW; ++t) {
#pragma unroll
    for (int v = 0; v < 8; ++v) {
      const int r = rowBase + t * 16 + v + 8 * half;  // C layout: VGPR v -> M = v + 8*half
      if (r < M) h[(size_t)r * 128 + n0 + l16] = c[t][v];
    }
  }
}

// ------------------- irregular-graph helper kernels ----------------------

__global__ void pdn_deg_init_kernel(float* __restrict__ deg, int N) {
  const int i = blockIdx.x * blockDim.x + threadIdx.x;
  if (i < N) deg[i] = 1.0f;  // self-loop fill weight
}

__global__ void pdn_deg_scatter_kernel(const int* __restrict__ col,
                                       const float* __restrict__ g,
                                       float* __restrict__ deg, int E) {
  const int e = blockIdx.x * blockDim.x + threadIdx.x;
  if (e < E) atomicAdd(&deg[col[e]], g[e]);
}

__global__ void pdn_dinv_kernel(float* __restrict__ deg, int N) {
  const int i = blockIdx.x * blockDim.x + threadIdx.x;
  if (i < N) {
    const float d = deg[i];
    deg[i] = (d > 0.0f) ? rsqrtf(d) : 0.0f;  // in place: deg -> dinv
  }
}

// out[i,:] = dinv[i]^2 * h[i,:]   (self-loop contribution, also initializes out)
__global__ void pdn_out_init_kernel(const float* __restrict__ h,
                                    const float* __restrict__ dinv,
                                    float* __restrict__ out, int N) {
  const int t = blockIdx.x * blockDim.x + threadIdx.x;
  const int i = t >> 7;
  if (i < N) {
    const float di = dinv[i];
    out[t] = di * di * h[t];
  }
}

// Per edge: out[col,:] += dinv[row]*g*dinv[col] * h[row,:]. One wave per edge,
// each lane handles a float4 (32*4 = 128 cols) -> 4 global_atomic_add_f32.
__global__ void pdn_agg_kernel(const float* __restrict__ h,
                               const int* __restrict__ row,
                               const int* __restrict__ col,
                               const float* __restrict__ g,
                               const float* __restrict__ dinv,
                               float* __restrict__ out, int E) {
  const int t = blockIdx.x * blockDim.x + threadIdx.x;
  const int e = t >> 5;
  const int lane = t & 31;
  if (e >= E) return;
  const int r = row[e];
  const int c = col[e];
  const float norm = dinv[r] * g[e] * dinv[c];
  const float4 hv = ((const float4*)(h + (size_t)r * 128))[lane];
  float* o = out + (size_t)c * 128 + lane * 4;
  atomicAdd(o + 0, norm * hv.x);
  atomicAdd(o + 1, norm * hv.y);
  atomicAdd(o + 2, norm * hv.z);
  atomicAdd(o + 3, norm * hv.w);
}

// -------------------------------------------------------------------------

extern "C" void kernel_launch(void* const* d_in, const int* in_sizes, int n_in,
                              void* d_out, int out_size, void* d_ws, size_t ws_size,
                              hipStream_t stream) {
  const float* x    = (const float*)d_in[0];
  const int*   ei   = (const int*)d_in[1];
  const float* ea   = (const float*)d_in[2];
  const float* W1   = (const float*)d_in[3];
  const float* m1w1 = (const float*)d_in[4];
  const float* m1b1 = (const float*)d_in[5];
  const float* m1w2 = (const float*)d_in[6];
  const float* m1b2 = (const float*)d_in[7];
  const float* W2   = (const float*)d_in[8];
  const float* m2w1 = (const float*)d_in[9];
  const float* m2b1 = (const float*)d_in[10];
  const float* m2w2 = (const float*)d_in[11];
  const float* m2b2 = (const float*)d_in[12];

  const int N = in_sizes[0] / 128;   // 100000
  const int E = in_sizes[2] / 16;    // 1600000
  const int* row = ei;       // edge_index[0] = source
  const int* col = ei + E;   // edge_index[1] = target

  float* ws    = (float*)d_ws;
  float* h_buf = ws;                                  // N*128
  float* a_buf = h_buf + (size_t)N * 128;             // N*128
  float* g_buf = a_buf + (size_t)N * 128;             // E
  float* deg   = g_buf + (size_t)E;                   // N (deg, then dinv)
  float* outp  = (float*)d_out;

  const int gateBlocks = (E + 127) / 128;          // 8 waves * 16 edges / block
  const int gemmBlocks = (N + 16 * RPW - 1) / (16 * RPW);
  const int nBlocks    = (N + 255) / 256;
  const int eBlocks    = (E + 255) / 256;
  const int neBlocks   = ((int)((size_t)N * 128 + 255)) / 256;
  const int aggBlocks  = (int)(((size_t)E * 32 + 255) / 256);

  // ---------------- Layer 1 ----------------
  pdn_gate_kernel<<<gateBlocks, 256, 0, stream>>>(ea, m1w1, m1b1, m1w2, m1b2, g_buf, E);
  pdn_deg_init_kernel<<<nBlocks, 256, 0, stream>>>(deg, N);
  pdn_deg_scatter_kernel<<<eBlocks, 256, 0, stream>>>(col, g_buf, deg, E);
  pdn_dinv_kernel<<<nBlocks, 256, 0, stream>>>(deg, N);
  pdn_gemm_kernel<<<gemmBlocks, 256, 0, stream>>>(x, W1, h_buf, N, 0);
  pdn_out_init_kernel<<<neBlocks, 256, 0, stream>>>(h_buf, deg, a_buf, N);
  pdn_agg_kernel<<<aggBlocks, 256, 0, stream>>>(h_buf, row, col, g_buf, deg, a_buf, E);

  // ---------------- Layer 2 (relu fused into GEMM A-load) ----------------
  pdn_gate_kernel<<<gateBlocks, 256, 0, stream>>>(ea, m2w1, m2b1, m2w2, m2b2, g_buf, E);
  pdn_deg_init_kernel<<<nBlocks, 256, 0, stream>>>(deg, N);
  pdn_deg_scatter_kernel<<<eBlocks, 256, 0, stream>>>(col, g_buf, deg, E);
  pdn_dinv_kernel<<<nBlocks, 256, 0, stream>>>(deg, N);
  pdn_gemm_kernel<<<gemmBlocks, 256, 0, stream>>>(a_buf, W2, h_buf, N, 1);
  pdn_out_init_kernel<<<neBlocks, 256, 0, stream>>>(h_buf, deg, outp, N);
  pdn_agg_kernel<<<aggBlocks, 256, 0, stream>>>(h_buf, row, col, g_buf, deg, outp, E);
}